// C3DLossKnn_15109694947701
// MI455X (gfx1250) — compile-verified
//
#include <hip/hip_runtime.h>
#include <hip/hip_bf16.h>
#include <math.h>

// ---------------------------------------------------------------------------
// C3D KNN kernel loss for MI455X (gfx1250, wave32).
// Distance matrix tiles via V_WMMA_F32_16X16X4_F32 (K=3 padded to 4).
// Reference chunk staged in LDS pre-swizzled into the WMMA B-operand lane
// layout so the hot loop is: ds_load_b64 (B) -> 2x wmma -> ds scatter ->
// s_wait_dscnt -> 8x ds_load_b128 row scan -> guarded top-20 insertion.
// ---------------------------------------------------------------------------

#define NPTS   8192
#define NBATCH 2
#define KNN    20
#define CHUNK  2048          // reference points staged in LDS per round
#define QT     256           // queries per block
#define NTERMS 4
#define ELLC   0.15f         // ell_min + ell_rand
#define BASED  10.0f
#define BIGF   1e10f
#define TOPINF 1e30f

typedef __attribute__((ext_vector_type(2))) float v2f;
typedef __attribute__((ext_vector_type(4))) float v4f;
typedef __attribute__((ext_vector_type(8))) float v8f;

__global__ __launch_bounds__(QT)
void c3d_knn_kernel(
    const float* __restrict__ g1p, const float* __restrict__ g1f, const float* __restrict__ g1n, const int* __restrict__ g1l,
    const float* __restrict__ p1p, const float* __restrict__ p1f, const float* __restrict__ p1n, const int* __restrict__ p1l,
    const float* __restrict__ f1p, const float* __restrict__ f1f, const float* __restrict__ f1n, const int* __restrict__ f1l,
    const float* __restrict__ g2p, const float* __restrict__ g2f, const float* __restrict__ g2n, const int* __restrict__ g2l,
    const float* __restrict__ p2p, const float* __restrict__ p2f, const float* __restrict__ p2n, const int* __restrict__ p2l,
    const float* __restrict__ f2p, const float* __restrict__ f2f, const float* __restrict__ f2n, const int* __restrict__ f2l,
    float* __restrict__ acc_out)
{
    // packed B operands: per 16-ref tile, 32 lanes x v2f (64 floats) -> 32 KB
    __shared__ float s_b[(CHUNK / 16) * 64];
    __shared__ float s_rr[CHUNK];                                        //  8 KB
    __shared__ float s_qx[QT], s_qy[QT], s_qz[QT];                       //  3 KB
    __shared__ float s_d[8][2 * 16 * 16];                                // 16 KB (per-wave dot tiles)
    __shared__ float s_red[QT];                                          //  1 KB

    const int tid   = threadIdx.x;
    const int lane  = tid & 31;
    const int wave  = tid >> 5;
    const int term  = blockIdx.z;   // 0..3
    const int batch = blockIdx.y;   // 0..1
    const int qbase = blockIdx.x * QT;

    // term -> (query cloud, reference cloud)
    const bool useG2 = (term >= 2);
    const float* q_pts = useG2 ? g2p : g1p;
    const float* q_ft  = useG2 ? g2f : g1f;
    const float* q_nrm = useG2 ? g2n : g1n;
    const int*   q_len = useG2 ? g2l : g1l;
    const float* r_pts = (term == 0) ? p1p : (term == 1) ? f1p : (term == 2) ? p2p : f2p;
    const float* r_ft  = (term == 0) ? p1f : (term == 1) ? f1f : (term == 2) ? p2f : f2f;
    const float* r_nrm = (term == 0) ? p1n : (term == 1) ? f1n : (term == 2) ? p2n : f2n;
    const int*   r_len = (term == 0) ? p1l : (term == 1) ? f1l : (term == 2) ? p2l : f2l;

    const size_t boff3 = (size_t)batch * NPTS * 3;
    const size_t boff4 = (size_t)batch * NPTS * 4;
    q_pts += boff3; q_ft += boff4; q_nrm += boff3;
    r_pts += boff3; r_ft += boff4; r_nrm += boff3;
    const int glen = q_len[batch];
    const int rlen = r_len[batch];

    __builtin_prefetch(r_pts + (size_t)tid * 3, 0, 1);   // global_prefetch_b8

    // ---- load own query (one query per thread) --------------------------------
    const int   qi  = qbase + tid;
    const float qx  = q_pts[(size_t)qi * 3 + 0];
    const float qy  = q_pts[(size_t)qi * 3 + 1];
    const float qz  = q_pts[(size_t)qi * 3 + 2];
    const float q2own = qx * qx + qy * qy + qz * qz;
    const float qh   = q_ft[(size_t)qi * 4 + 0];
    const float qs   = q_ft[(size_t)qi * 4 + 1];
    const float qv   = q_ft[(size_t)qi * 4 + 2];
    const float qres = q_ft[(size_t)qi * 4 + 3];
    const float qnx  = q_nrm[(size_t)qi * 3 + 0];
    const float qny  = q_nrm[(size_t)qi * 3 + 1];
    const float qnz  = q_nrm[(size_t)qi * 3 + 2];

    s_qx[tid] = qx; s_qy[tid] = qy; s_qz[tid] = qz;
    __syncthreads();

    // ---- WMMA A operands (16x4 f32 layout: lanes 0-15 K=0,1; lanes 16-31 K=2,3)
    const bool lo   = (lane < 16);
    const int  mrow = wave * 32 + (lane & 15);
    v2f a0, a1;
    a0.x = lo ? s_qx[mrow]      : s_qz[mrow];
    a0.y = lo ? s_qy[mrow]      : 0.0f;
    a1.x = lo ? s_qx[mrow + 16] : s_qz[mrow + 16];
    a1.y = lo ? s_qy[mrow + 16] : 0.0f;

    // ---- per-lane top-K (sorted ascending) ------------------------------------
    float kd[KNN];
    int   ki[KNN];
#pragma unroll
    for (int j = 0; j < KNN; ++j) { kd[j] = TOPINF; ki[j] = 0; }

    float* dt = &s_d[wave][0];
    const int hi   = lane >> 4;
    const int colw = lane & 15;
    // row this lane owns in its wave's dot tiles (query = wave*32 + lane)
    float* myrow = dt + hi * 256 + colw * 16;

    // ---- stream reference cloud ----------------------------------------------
    for (int cb = 0; cb < NPTS; cb += CHUNK) {
        __syncthreads();   // previous chunk fully consumed
        for (int j = tid; j < CHUNK; j += QT) {
            const int gi = cb + j;
            const float rx = r_pts[(size_t)gi * 3 + 0];
            const float ry = r_pts[(size_t)gi * 3 + 1];
            const float rz = r_pts[(size_t)gi * 3 + 2];
            // pre-swizzle into B-operand lane layout for tile j/16:
            // lane l = j%16 holds (x,y) = rows K0,K1; lane 16+l holds (z,0) = rows K2,K3
            float* bb = &s_b[(j >> 4) * 64];
            const int l = j & 15;
            v2f blo; blo.x = rx; blo.y = ry;
            v2f bhi; bhi.x = rz; bhi.y = 0.0f;
            *(v2f*)(bb + l * 2)        = blo;   // ds_store_b64
            *(v2f*)(bb + (16 + l) * 2) = bhi;   // ds_store_b64
            s_rr[j] = (gi < rlen) ? (rx * rx + ry * ry + rz * rz) : BIGF;
        }
        __syncthreads();

        for (int nt = 0; nt < CHUNK; nt += 16) {
            // B operand: one uniform ds_load_b64 per lane, EXEC stays all-1s
            const v2f b = *(const v2f*)(&s_b[(nt >> 4) * 64] + lane * 2);

            v8f c0 = {}; v8f c1 = {};
            c0 = __builtin_amdgcn_wmma_f32_16x16x4_f32(false, a0, false, b,
                                                       (short)0, c0, false, false);
            c1 = __builtin_amdgcn_wmma_f32_16x16x4_f32(false, a1, false, b,
                                                       (short)0, c1, false, false);

            // scatter accumulators to LDS: C layout -> element (m,n): m = v + 8*hi, n = colw
#pragma unroll
            for (int v = 0; v < 8; ++v) {
                dt[        (v + 8 * hi) * 16 + colw] = c0[v];
                dt[256 +   (v + 8 * hi) * 16 + colw] = c1[v];
            }
            __builtin_amdgcn_wave_barrier();
            asm volatile("s_wait_dscnt 0" ::: "memory");
            __builtin_amdgcn_wave_barrier();

            // each lane scans its own query row: d = q2 + r2[col] - 2*dot
            // vectorized: 4x ds_load_b128 (dots) + 4x ds_load_b128 (r2, broadcast)
            const v4f* rowv = (const v4f*)myrow;
            const v4f* rrv  = (const v4f*)(&s_rr[nt]);
#pragma unroll
            for (int cc = 0; cc < 4; ++cc) {
                const v4f dot4 = rowv[cc];
                const v4f rr4  = rrv[cc];
#pragma unroll
                for (int u = 0; u < 4; ++u) {
                    const float d = q2own + rr4[u] - 2.0f * dot4[u];
                    if (d < kd[KNN - 1]) {
                        float cd = d; int ci = cb + nt + cc * 4 + u;
#pragma unroll
                        for (int j = 0; j < KNN; ++j) {
                            const bool  sw = cd < kd[j];
                            const float od = kd[j]; const int oi = ki[j];
                            kd[j] = sw ? cd : od;   ki[j] = sw ? ci : oi;
                            cd    = sw ? od : cd;   ci    = sw ? oi : ci;
                        }
                    }
                }
            }
            __builtin_amdgcn_wave_barrier();  // keep next tile's stores behind the reads
        }
    }

    // ---- per-neighbor kernels -------------------------------------------------
    float acc = 0.0f;
    if (qi < glen) {
        float lsv = ELLC * (qz - BASED) / BASED;
        lsv = fmaxf(lsv, ELLC);
        const float inv_ls = 1.0f / (lsv * lsv);
#pragma unroll 4
        for (int k = 0; k < KNN; ++k) {
            const int i = ki[k];
            const float dist_k = __expf(-kd[k] * inv_ls);
            const float rh  = r_ft[(size_t)i * 4 + 0];
            const float rs  = r_ft[(size_t)i * 4 + 1];
            const float rv  = r_ft[(size_t)i * 4 + 2];
            const float rre = r_ft[(size_t)i * 4 + 3];
            const float dh = qh - rh, ds = qs - rs, dv = qv - rv;
            const float cdist   = sqrtf(dh * dh + ds * ds + dv * dv);
            const float color_k = __expf(-cdist * 5.0f);            // /COLOR_SCALE(0.2)
            const float alpha   = 0.2f / (0.1f + qres + rre);       // 2*RM_MIN/(2*RM_MIN/RM_MAX + ...)
            const float rnx = r_nrm[(size_t)i * 3 + 0];
            const float rny = r_nrm[(size_t)i * 3 + 1];
            const float rnz = r_nrm[(size_t)i * 3 + 2];
            const float ndot = qnx * rnx + qny * rny + qnz * rnz;
            const float normal_k = fmaxf(ndot * alpha, 0.0f);
            acc += dist_k * color_k * normal_k;
        }
    }

    // ---- block reduction + global accumulate ----------------------------------
    s_red[tid] = acc;
    __syncthreads();
    for (int s = QT / 2; s > 0; s >>= 1) {
        if (tid < s) s_red[tid] += s_red[tid + s];
        __syncthreads();
    }
    if (tid == 0) atomicAdd(&acc_out[term * NBATCH + batch], s_red[0]);
}

__global__ void c3d_init_ws(float* __restrict__ ws)
{
    if (threadIdx.x < NTERMS * NBATCH) ws[threadIdx.x] = 0.0f;
}

__global__ void c3d_finalize(const float* __restrict__ ws,
                             const int* __restrict__ len1,
                             const int* __restrict__ len2,
                             float* __restrict__ out)
{
    if (threadIdx.x == 0) {
        float loss = 0.0f;
        for (int b = 0; b < NBATCH; ++b) {
            const float inv1 = 1.0f / (float)len1[b];
            const float inv2 = 1.0f / (float)len2[b];
            const float s = ws[0 * NBATCH + b] * inv1 + ws[1 * NBATCH + b] * inv1 +
                            ws[2 * NBATCH + b] * inv2 + ws[3 * NBATCH + b] * inv2;
            loss += 0.25f * s;
        }
        out[0] = -loss / (float)NBATCH;
    }
}

extern "C" void kernel_launch(void* const* d_in, const int* in_sizes, int n_in,
                              void* d_out, int out_size, void* d_ws, size_t ws_size,
                              hipStream_t stream)
{
    (void)in_sizes; (void)n_in; (void)out_size; (void)ws_size;
    // setup_inputs() order: for each cloud {points, feat, normals, len}
    // clouds: gt_1, pred_1, flowed_1_from_2, gt_2, pred_2, flowed_2_from_1
    const float* g1p = (const float*)d_in[0];  const float* g1f = (const float*)d_in[1];
    const float* g1n = (const float*)d_in[2];  const int*   g1l = (const int*)  d_in[3];
    const float* p1p = (const float*)d_in[4];  const float* p1f = (const float*)d_in[5];
    const float* p1n = (const float*)d_in[6];  const int*   p1l = (const int*)  d_in[7];
    const float* f1p = (const float*)d_in[8];  const float* f1f = (const float*)d_in[9];
    const float* f1n = (const float*)d_in[10]; const int*   f1l = (const int*)  d_in[11];
    const float* g2p = (const float*)d_in[12]; const float* g2f = (const float*)d_in[13];
    const float* g2n = (const float*)d_in[14]; const int*   g2l = (const int*)  d_in[15];
    const float* p2p = (const float*)d_in[16]; const float* p2f = (const float*)d_in[17];
    const float* p2n = (const float*)d_in[18]; const int*   p2l = (const int*)  d_in[19];
    const float* f2p = (const float*)d_in[20]; const float* f2f = (const float*)d_in[21];
    const float* f2n = (const float*)d_in[22]; const int*   f2l = (const int*)  d_in[23];

    float* ws = (float*)d_ws;

    c3d_init_ws<<<1, 32, 0, stream>>>(ws);

    dim3 grid(NPTS / QT, NBATCH, NTERMS);   // (32, 2, 4)
    c3d_knn_kernel<<<grid, QT, 0, stream>>>(
        g1p, g1f, g1n, g1l,  p1p, p1f, p1n, p1l,  f1p, f1f, f1n, f1l,
        g2p, g2f, g2n, g2l,  p2p, p2f, p2n, p2l,  f2p, f2f, f2n, f2l,
        ws);

    c3d_finalize<<<1, 32, 0, stream>>>(ws, g1l, g2l, (float*)d_out);
}